// Model_5815385719290
// MI455X (gfx1250) — compile-verified
//
#include <hip/hip_runtime.h>

typedef float v2f __attribute__((ext_vector_type(2)));
typedef float v8f __attribute__((ext_vector_type(8)));

#define Vn 16
#define Cn 16
#define Hn 1024
#define Wn 1024
#define RMn 8
#define RDn 4
#define PX_PER_BLOCK 128   // 8 waves * 16 pixels

struct cpx { float re, im; };

__device__ __forceinline__ cpx cmul(cpx a, cpx b) {
    return { a.re * b.re - a.im * b.im, a.re * b.im + a.im * b.re };
}

__device__ __forceinline__ cpx load2(const float* __restrict__ p, int idx) {
    float2 t = *(const float2*)(p + 2 * (size_t)idx);
    return { t.x, t.y };
}

// fast atan2 via minimax atan poly on [0,1] + octant fixup (HW has no atan)
__device__ __forceinline__ float fast_atan2f(float y, float x) {
    float ax = fabsf(x), ay = fabsf(y);
    float mx = fmaxf(ax, ay), mn = fminf(ax, ay);
    float a  = __fdividef(mn, mx);
    float s  = a * a;
    float q  = fmaf(s, 0.0208351f, -0.0851330f);
    q = fmaf(s, q, 0.1801410f);
    q = fmaf(s, q, -0.3302995f);
    q = fmaf(s, q, 0.9998660f);
    float r = a * q;
    if (ay > ax)  r = 1.57079632679f - r;
    if (x < 0.0f) r = 3.14159265359f - r;
    return copysignf(r, y);
}

__device__ __forceinline__ cpx clog_fast(cpx z) {
    float m2 = fmaf(z.re, z.re, z.im * z.im);
    return { 0.5f * __logf(m2), fast_atan2f(z.im, z.re) };
}

// exp(w * l) where l = log(z)
__device__ __forceinline__ cpx cexp_wl(cpx w, cpx l) {
    cpx t = cmul(w, l);
    float m = __expf(t.re);
    float sn, cs;
    __sincosf(t.im, &sn, &cs);
    return { m * cs, m * sn };
}

__global__ __launch_bounds__(256)
void powsum_wmma_kernel(const float* __restrict__ base,
                        const float* __restrict__ mod_u,
                        const float* __restrict__ mod_v,
                        const float* __restrict__ del_u,
                        const float* __restrict__ del_v,
                        const float* __restrict__ exp_base,
                        const float* __restrict__ exp_mod,
                        const float* __restrict__ exp_del,
                        const float* __restrict__ bias,
                        const float* __restrict__ scale,
                        const float* __restrict__ coef,
                        float* __restrict__ out)
{
    const int wave = threadIdx.x >> 5;
    const int lane = threadIdx.x & 31;
    const int p    = lane & 15;       // pixel within tile / N column / A row m
    const int hi   = lane >> 4;       // half-wave select
    const int h    = blockIdx.y;
    const int w    = blockIdx.x * PX_PER_BLOCK + wave * 16 + p;

    // ---- shared-per-pixel base power: pb = exp(exp_base * log(base)) ----
    cpx b  = load2(base, h * Wn + w);
    cpx lb = clog_fast(b);
    cpx eb = { exp_base[0], exp_base[1] };
    cpx pb = cexp_wl(eb, lb);

    // ---- per-lane: comp for 8 (c, pixel) pairs, laid out as WMMA B tiles ----
    // B vgpr j of K-chunk kk holds comp[c = 4*kk + j + 2*hi][pixel p]
    float Br[4][2], Bi[4][2];
#pragma unroll
    for (int j = 0; j < 8; ++j) {
        const int kk  = j >> 1;
        const int idx = j & 1;
        const int c   = kk * 4 + idx + 2 * hi;

        // rank-8 complex reconstruction: mod = sum_r mod_u[c,h,r] * mod_v[c,r,w]
        cpx m = { 0.0f, 0.0f };
#pragma unroll
        for (int r = 0; r < RMn; ++r) {
            cpx u = load2(mod_u, (c * Hn + h) * RMn + r);
            cpx v = load2(mod_v, (c * RMn + r) * Wn + w);
            m.re = fmaf(u.re, v.re, fmaf(-u.im, v.im, m.re));
            m.im = fmaf(u.re, v.im, fmaf( u.im, v.re, m.im));
        }
        // rank-4 complex reconstruction
        cpx d = { 0.0f, 0.0f };
#pragma unroll
        for (int r = 0; r < RDn; ++r) {
            cpx u = load2(del_u, (c * Hn + h) * RDn + r);
            cpx v = load2(del_v, (c * RDn + r) * Wn + w);
            d.re = fmaf(u.re, v.re, fmaf(-u.im, v.im, d.re));
            d.im = fmaf(u.re, v.im, fmaf( u.im, v.re, d.im));
        }

        cpx pm = cexp_wl(load2(exp_mod, c), clog_fast(m));
        cpx pd = cexp_wl(load2(exp_del, c), clog_fast(d));

        cpx q = cmul(pb, pm);
        q.re += pd.re;
        q.im += pd.im;
        cpx sc = load2(scale, c);
        cpx bs = load2(bias, c);
        cpx comp = cmul(sc, q);
        comp.re += bs.re;
        comp.im += bs.im;

        Br[kk][idx] = comp.re;
        Bi[kk][idx] = comp.im;
    }

    // ---- complex 16x16 GEMM over C via V_WMMA_F32_16X16X4_F32 (4 K-chunks) ----
    // outR = coefR*compR - coefI*compI ; outI = coefR*compI + coefI*compR
    v8f accR = {};
    v8f accI = {};
#pragma unroll
    for (int kk = 0; kk < 4; ++kk) {
        const int k0 = kk * 4 + 2 * hi;           // A vgpr j holds coef[p][k0 + j]
        cpx a0 = load2(coef, p * Cn + k0);
        cpx a1 = load2(coef, p * Cn + k0 + 1);
        v2f Ar  = {  a0.re,  a1.re };
        v2f Ai  = {  a0.im,  a1.im };
        v2f AiN = { -a0.im, -a1.im };
        v2f BrV = { Br[kk][0], Br[kk][1] };
        v2f BiV = { Bi[kk][0], Bi[kk][1] };

        accR = __builtin_amdgcn_wmma_f32_16x16x4_f32(false, Ar,  false, BrV, (short)0, accR, false, false);
        accR = __builtin_amdgcn_wmma_f32_16x16x4_f32(false, AiN, false, BiV, (short)0, accR, false, false);
        accI = __builtin_amdgcn_wmma_f32_16x16x4_f32(false, Ar,  false, BiV, (short)0, accI, false, false);
        accI = __builtin_amdgcn_wmma_f32_16x16x4_f32(false, Ai,  false, BrV, (short)0, accI, false, false);
    }

    // ---- store: D vgpr r -> v = r + 8*hi, column p; out is [V,H,W,2] ----
#pragma unroll
    for (int r = 0; r < 8; ++r) {
        const int v = r + 8 * hi;
        float2 o;
        o.x = accR[r];
        o.y = accI[r];
        *(float2*)(out + ((size_t)(v * Hn + h) * Wn + w) * 2) = o;
    }
}

extern "C" void kernel_launch(void* const* d_in, const int* in_sizes, int n_in,
                              void* d_out, int out_size, void* d_ws, size_t ws_size,
                              hipStream_t stream) {
    (void)in_sizes; (void)n_in; (void)d_ws; (void)ws_size; (void)out_size;
    const float* base     = (const float*)d_in[0];
    const float* mod_u    = (const float*)d_in[1];
    const float* mod_v    = (const float*)d_in[2];
    const float* del_u    = (const float*)d_in[3];
    const float* del_v    = (const float*)d_in[4];
    const float* exp_base = (const float*)d_in[5];
    const float* exp_mod  = (const float*)d_in[6];
    const float* exp_del  = (const float*)d_in[7];
    const float* bias     = (const float*)d_in[8];
    const float* scale    = (const float*)d_in[9];
    const float* coef     = (const float*)d_in[10];
    float* out = (float*)d_out;

    dim3 grid(Wn / PX_PER_BLOCK, Hn, 1);   // (8, 1024)
    dim3 block(256, 1, 1);                 // 8 waves, one 16-px WMMA tile each
    powsum_wmma_kernel<<<grid, block, 0, stream>>>(
        base, mod_u, mod_v, del_u, del_v,
        exp_base, exp_mod, exp_del, bias, scale, coef, out);
}